// Decoder_61194694033537
// MI455X (gfx1250) — compile-verified
//
#include <hip/hip_runtime.h>
#include <hip/hip_bf16.h>

// ---------------------------------------------------------------------------
// Decoder layer for MI455X (gfx1250), f32 throughout via V_WMMA_F32_16X16X4_F32
//   B=8 S=512 D=512 HEAD=64 DK=4 DV=8 HIDD=10
// ---------------------------------------------------------------------------

typedef float v2f __attribute__((ext_vector_type(2)));
typedef float v8f __attribute__((ext_vector_type(8)));

#define D_MODEL 512
#define SEQ     512
#define BATCH   8
#define NHEAD   64
#define DK      4
#define DV      8
#define HIDD    10
#define ROWS    (BATCH * SEQ)   // 4096

static __device__ __forceinline__ v8f wmma_f32(v2f a, v2f b, v8f c) {
  // D = A(16x4) * B(4x16) + C(16x16), all f32.
  return __builtin_amdgcn_wmma_f32_16x16x4_f32(false, a, false, b, (short)0, c,
                                               false, false);
}

// ---- async global->LDS copy (gfx1250 GLOBAL_LOAD_ASYNC_TO_LDS_B128), guarded
#if __has_builtin(__builtin_amdgcn_global_load_async_to_lds_b128) && \
    __has_builtin(__builtin_amdgcn_s_wait_asynccnt)
#define HAVE_ASYNC_LDS 1
typedef int g4i __attribute__((vector_size(16)));              // generic V4i
typedef __attribute__((address_space(1))) g4i* g4i_gptr;       // global
typedef __attribute__((address_space(3))) g4i* g4i_lptr;       // LDS
static __device__ __forceinline__ void async_cp16(void* lds_dst, const void* gsrc) {
  __builtin_amdgcn_global_load_async_to_lds_b128((g4i_gptr)gsrc, (g4i_lptr)lds_dst,
                                                 0, 0);
}
static __device__ __forceinline__ void async_wait() {
  __builtin_amdgcn_s_wait_asynccnt(0);
}
#else
#define HAVE_ASYNC_LDS 0
static __device__ __forceinline__ void async_cp16(void* lds_dst, const void* gsrc) {
  *(float4*)lds_dst = *(const float4*)gsrc;
}
static __device__ __forceinline__ void async_wait() {}
#endif

// ---------------------------------------------------------------------------
// GEMM: Out[M x N] = X[M x 512] @ W[512 x N], M=4096, N in {256, 512}
// grid = (M/16, N/64), block = 128 (4 waves); wave w -> N-tile blockIdx.y*64+16w
// Both operands staged in LDS; inner loop is ds_load + v_wmma only.
// ---------------------------------------------------------------------------
#define TILE_K 64
#define APAD   68   // row stride (floats): 16B-aligned rows, conflict-free frags

__global__ __launch_bounds__(128)
void gemm_xw(const float* __restrict__ X, const float* __restrict__ W,
             float* __restrict__ Out, int N) {
  __shared__ float As[16][APAD];        // 16 x 64 A slab
  __shared__ float Bs[TILE_K][APAD];    // 64(k) x 64(n) W chunk
  const int m0   = blockIdx.x * 16;
  const int wv   = threadIdx.x >> 5;
  const int lane = threadIdx.x & 31;
  const int hl   = lane >> 4;           // half-wave select (k +0/+2)
  const int l16  = lane & 15;
  const int nb0  = blockIdx.y * 64;
  const int cb   = wv * 16 + l16;       // this wave's column inside the chunk

  v8f acc = {};
  for (int k0 = 0; k0 < D_MODEL; k0 += TILE_K) {
    // stage A: 16 rows x 64 cols, 16B per copy, coalesced
    for (int i = threadIdx.x; i < 16 * (TILE_K / 4); i += 128) {
      const int r = i >> 4, c4 = (i & 15) * 4;
      async_cp16(&As[r][c4], &X[(m0 + r) * D_MODEL + k0 + c4]);
    }
    // stage W chunk: 64 rows x 64 cols, 16B per copy, coalesced
    for (int i = threadIdx.x; i < TILE_K * (TILE_K / 4); i += 128) {
      const int kr = i >> 4, c4 = (i & 15) * 4;
      async_cp16(&Bs[kr][c4], &W[(k0 + kr) * N + nb0 + c4]);
    }
    if (k0 + TILE_K < D_MODEL)
      __builtin_prefetch(&X[(m0 + (threadIdx.x & 15)) * D_MODEL + k0 + TILE_K], 0, 3);
    async_wait();
    __syncthreads();

#pragma unroll
    for (int kk = 0; kk < TILE_K; kk += 4) {
      const int ka = kk + hl * 2;
      // A fragment: lanes 0-15 -> M=l16,K={kk,kk+1}; lanes 16-31 -> K={kk+2,kk+3}
      v2f a = *(const v2f*)&As[l16][ka];
      // B fragment: B[k][n] = Bs[ka..ka+1][cb]
      v2f b;
      b.x = Bs[ka][cb];
      b.y = Bs[ka + 1][cb];
      acc = wmma_f32(a, b, acc);
    }
    __syncthreads();
  }
  // D layout: VGPR r -> M = r (+8 for upper half), N = l16
#pragma unroll
  for (int r = 0; r < 8; ++r)
    Out[(m0 + r + hl * 8) * N + nb0 + wv * 16 + l16] = acc[r];
}

// ---------------------------------------------------------------------------
// Attention, one block per (b,h); flash-style streaming softmax that keeps the
// reference's multiplicative tril-mask quirk (masked scores are 0, and exp(0)
// still participates in the softmax).
// grid = B*NHEAD = 512, block = 128 (4 waves); wave handles q-tiles wv,wv+4,...
// ---------------------------------------------------------------------------
__global__ __launch_bounds__(128)
void attn(const float* __restrict__ Q, const float* __restrict__ Kx,
          const float* __restrict__ V, float* __restrict__ O, int causal) {
  __shared__ float Ks[SEQ][DK];        //  8 KB
  __shared__ float Vs[SEQ][DV];        // 16 KB
  __shared__ float Ps[4][16][18];      //  ~4.5 KB, padded stride 18

  const int bh   = blockIdx.x;
  const int b    = bh >> 6;
  const int h    = bh & 63;
  const int wv   = threadIdx.x >> 5;
  const int lane = threadIdx.x & 31;
  const int hl   = lane >> 4;
  const int l16  = lane & 15;

  // stage K (512x4) and V (512x8) for this (b,h) -- async b128 copies
  for (int s = threadIdx.x; s < SEQ; s += 128) {
    const float* kp = &Kx[((b * SEQ + s) * NHEAD + h) * DK];
    async_cp16(&Ks[s][0], kp);
    const float* vp = &V[((b * SEQ + s) * NHEAD + h) * DV];
    async_cp16(&Vs[s][0], vp);
    async_cp16(&Vs[s][4], vp + 4);
  }
  async_wait();
  __syncthreads();

  for (int qt = wv; qt < SEQ / 16; qt += 4) {
    const int q0 = qt * 16;
    // Q fragment (A-layout, K=DK=4 exactly -> one frag for the whole row)
    v2f aq = *(const v2f*)&Q[((b * SEQ + q0 + l16) * NHEAD + h) * DK + hl * 2];

    float m_run[8], l_run[8];
#pragma unroll
    for (int r = 0; r < 8; ++r) { m_run[r] = -3.4e38f; l_run[r] = 0.f; }
    v8f oacc = {};

    for (int jt = 0; jt < SEQ / 16; ++jt) {
      const int j0 = jt * 16;
      // score tile: one f32 WMMA (16x16x4)
      v2f bk = *(const v2f*)&Ks[j0 + l16][hl * 2];
      v8f sc = {};
      sc = wmma_f32(aq, bk, sc);

      // scale by 1/sqrt(DK)=0.5 and apply MULTIPLICATIVE causal mask (quirk)
      const int jcol = j0 + l16;
#pragma unroll
      for (int r = 0; r < 8; ++r) {
        const int qrow = q0 + r + hl * 8;
        float s = sc[r] * 0.5f;
        if (causal && jcol > qrow) s = 0.0f;
        sc[r] = s;
      }

      // per-row online softmax update (rows live in 16-lane halves)
#pragma unroll
      for (int r = 0; r < 8; ++r) {
        float mx = sc[r];
        for (int off = 8; off > 0; off >>= 1)
          mx = fmaxf(mx, __shfl_xor(mx, off, 16));
        const float mnew  = fmaxf(m_run[r], mx);
        const float p     = __expf(sc[r] - mnew);
        float psum = p;
        for (int off = 8; off > 0; off >>= 1)
          psum += __shfl_xor(psum, off, 16);
        const float alpha = __expf(m_run[r] - mnew);
        l_run[r] = l_run[r] * alpha + psum;
        m_run[r] = mnew;
        oacc[r] *= alpha;
        Ps[wv][r + hl * 8][l16] = p;     // D-layout -> LDS for A-layout reload
      }

      // P(16x16) @ V-tile(16x8, padded to 16): four k=4 WMMA steps
#pragma unroll
      for (int kk = 0; kk < 16; kk += 4) {
        const int ka = kk + hl * 2;
        v2f ap = *(const v2f*)&Ps[wv][l16][ka];
        const int jb = j0 + ka;
        v2f bv;
        bv.x = (l16 < DV) ? Vs[jb][l16]     : 0.0f;
        bv.y = (l16 < DV) ? Vs[jb + 1][l16] : 0.0f;
        oacc = wmma_f32(ap, bv, oacc);
      }
    }

    // normalize and write out: col = h*DV + n, n = l16 (< 8 valid)
#pragma unroll
    for (int r = 0; r < 8; ++r) {
      const float ov = oacc[r] / l_run[r];
      if (l16 < DV)
        O[(b * SEQ + q0 + r + hl * 8) * D_MODEL + h * DV + l16] = ov;
    }
  }
}

// ---------------------------------------------------------------------------
// Residual + LayerNorm + ReLU MLP (512 -> 10 -> 512). One block per row.
// ---------------------------------------------------------------------------
__global__ __launch_bounds__(256)
void ln_ffn(const float* __restrict__ x, const float* __restrict__ fx,
            const float* __restrict__ g, const float* __restrict__ bt,
            const float* __restrict__ w1, const float* __restrict__ b1,
            const float* __restrict__ w2, const float* __restrict__ b2,
            float* __restrict__ out) {
  __shared__ float ys[D_MODEL];
  __shared__ float hs[HIDD];
  __shared__ float red[16];
  const int row = blockIdx.x;
  const int t   = threadIdx.x;

  const float v0 = x[row * D_MODEL + t]       + fx[row * D_MODEL + t];
  const float v1 = x[row * D_MODEL + t + 256] + fx[row * D_MODEL + t + 256];
  float s = v0 + v1, ss = v0 * v0 + v1 * v1;
  for (int off = 16; off > 0; off >>= 1) {
    s  += __shfl_xor(s,  off, 32);
    ss += __shfl_xor(ss, off, 32);
  }
  const int wv = t >> 5, lane = t & 31;
  if (lane == 0) { red[wv] = s; red[8 + wv] = ss; }
  __syncthreads();
  if (t == 0) {
    float S = 0.f, SS = 0.f;
    for (int i = 0; i < 8; ++i) { S += red[i]; SS += red[8 + i]; }
    red[0] = S; red[8] = SS;
  }
  __syncthreads();
  const float mu  = red[0] * (1.0f / D_MODEL);
  const float var = red[8] * (1.0f / D_MODEL) - mu * mu;
  const float rs  = rsqrtf(var + 1e-5f);
  const float y0 = (v0 - mu) * rs * g[t]       + bt[t];
  const float y1 = (v1 - mu) * rs * g[t + 256] + bt[t + 256];
  ys[t] = y0; ys[t + 256] = y1;
  __syncthreads();

  if (t < 32) {
    float acc[HIDD];
#pragma unroll
    for (int j = 0; j < HIDD; ++j) acc[j] = 0.f;
    for (int c = t; c < D_MODEL; c += 32) {
      const float yv = ys[c];
#pragma unroll
      for (int j = 0; j < HIDD; ++j) acc[j] += yv * w1[j * D_MODEL + c];
    }
#pragma unroll
    for (int j = 0; j < HIDD; ++j)
      for (int off = 16; off > 0; off >>= 1)
        acc[j] += __shfl_xor(acc[j], off, 32);
    if (t < HIDD) hs[t] = fmaxf(acc[t] + b1[t], 0.f);
  }
  __syncthreads();

  float o0 = b2[t], o1 = b2[t + 256];
#pragma unroll
  for (int j = 0; j < HIDD; ++j) {
    o0 += hs[j] * w2[t * HIDD + j];
    o1 += hs[j] * w2[(t + 256) * HIDD + j];
  }
  out[row * D_MODEL + t]       = o0;
  out[row * D_MODEL + t + 256] = o1;
}

// ---------------------------------------------------------------------------
extern "C" void kernel_launch(void* const* d_in, const int* in_sizes, int n_in,
                              void* d_out, int out_size, void* d_ws, size_t ws_size,
                              hipStream_t stream) {
  (void)in_sizes; (void)n_in; (void)out_size; (void)ws_size;
  const float* x     = (const float*)d_in[0];
  const float* x_enc = (const float*)d_in[1];
  const float* WqM   = (const float*)d_in[2];
  const float* WkM   = (const float*)d_in[3];
  const float* WvM   = (const float*)d_in[4];
  const float* ln_g  = (const float*)d_in[5];
  const float* ln_b  = (const float*)d_in[6];
  const float* fc1_w = (const float*)d_in[7];
  const float* fc1_b = (const float*)d_in[8];
  const float* fc2_w = (const float*)d_in[9];
  const float* fc2_b = (const float*)d_in[10];
  float* outp = (float*)d_out;

  // workspace layout (floats): Q/K (1M each), V (2M), fx (2M) -> 24 MB total
  float* Qb = (float*)d_ws;
  float* Kb = Qb + ROWS * (NHEAD * DK);
  float* Vb = Kb + ROWS * (NHEAD * DK);
  float* Fb = Vb + ROWS * (NHEAD * DV);

  const dim3 gQK(ROWS / 16, (NHEAD * DK) / 64);   // 256 x 4
  const dim3 gV (ROWS / 16, (NHEAD * DV) / 64);   // 256 x 8

  // self-attention projections
  gemm_xw<<<gQK, 128, 0, stream>>>(x, WqM, Qb, NHEAD * DK);
  gemm_xw<<<gQK, 128, 0, stream>>>(x, WkM, Kb, NHEAD * DK);
  gemm_xw<<<gV,  128, 0, stream>>>(x, WvM, Vb, NHEAD * DV);
  // self-attention (multiplicative causal mask quirk)
  attn<<<BATCH * NHEAD, 128, 0, stream>>>(Qb, Kb, Vb, Fb, 1);
  // cross-attention projections (reuse Q/K/V buffers; fx consumed first)
  gemm_xw<<<gQK, 128, 0, stream>>>(Fb,    WqM, Qb, NHEAD * DK);
  gemm_xw<<<gQK, 128, 0, stream>>>(x_enc, WkM, Kb, NHEAD * DK);
  gemm_xw<<<gV,  128, 0, stream>>>(x_enc, WvM, Vb, NHEAD * DV);
  // cross-attention (no mask); overwrites Fb (safe: old fx already consumed)
  attn<<<BATCH * NHEAD, 128, 0, stream>>>(Qb, Kb, Vb, Fb, 0);
  // residual + LN + MLP
  ln_ffn<<<ROWS, 256, 0, stream>>>(x, Fb, ln_g, ln_b, fc1_w, fc1_b,
                                   fc2_w, fc2_b, outp);
}